// SoftDPMatch_10127532884778
// MI455X (gfx1250) — compile-verified
//
#include <hip/hip_runtime.h>

#ifndef __has_builtin
#define __has_builtin(x) 0
#endif

// Async global->LDS path (gfx1250 ASYNCcnt pipeline). Guarded so compile
// succeeds on toolchains that lack the builtins (falls back to sync loads).
#if defined(__HIP_DEVICE_COMPILE__) && defined(__gfx1250__) && \
    __has_builtin(__builtin_amdgcn_global_load_async_to_lds_b32) && \
    __has_builtin(__builtin_amdgcn_s_wait_asynccnt)
#define DP_USE_ASYNC 1
#else
#define DP_USE_ASYNC 0
#endif

namespace {
constexpr int   Nn   = 512;
constexpr int   Mm   = 512;
constexpr int   KD   = Nn + Mm - 1;   // 1023 anti-diagonals
constexpr float NEGV = -1e30f;
constexpr float GAMMAc = 1.0f;
constexpr float Tc     = 1.0f;
}

#if DP_USE_ASYNC
typedef __attribute__((address_space(1))) int* gint_ptr;
typedef __attribute__((address_space(3))) int* lint_ptr;
#endif

__device__ __forceinline__ void dp_stage_load(const float* g, float* l) {
#if DP_USE_ASYNC
  __builtin_amdgcn_global_load_async_to_lds_b32(
      (gint_ptr)(g), (lint_ptr)(l),
      /*offset=*/0, /*cpol=*/0);
#else
  __builtin_prefetch(g + Mm, 0, 0);   // emits global_prefetch_b8
  *l = *g;
#endif
}

// Order-preserving float -> uint key (works for negatives), for atomic max.
__device__ __forceinline__ unsigned enc_f32(float x) {
  unsigned u = __float_as_uint(x);
  return (u & 0x80000000u) ? ~u : (u | 0x80000000u);
}
__device__ __forceinline__ float dec_f32(unsigned u) {
  return (u & 0x80000000u) ? __uint_as_float(u ^ 0x80000000u)
                           : __uint_as_float(~u);
}

// One workgroup per DP instance. Thread j owns DP column j (reversed column
// for BACK). Anti-diagonal wavefront: triple-buffered R diagonals in LDS,
// 2-deep async-staged D double buffer (issue diag k+1, s_wait_asynccnt 1,
// consume diag k).
//
//  BACK=0            : T[i,j]  = f DP value
//  BACK=1, COMBINE=0 : T[r,c]  = b DP value (raw)
//  BACK=1, COMBINE=1 : T[r,c] <- T[r,c] + b - d  (logit, in place) + gmax
template <bool BACK, bool COMBINE>
__device__ __forceinline__ void dp_core(const float* __restrict__ D,
                                        float* __restrict__ T,
                                        unsigned* __restrict__ gmax,
                                        int batch) {
  const int j = threadIdx.x;                       // DP column (0..511)
  const size_t base = (size_t)batch * Nn * Mm;
  const float* Db = D + base;
  float*       Tb = T + base;

  __shared__ float rbuf[3][Mm];     // R on diagonals k, k-1, k-2
  __shared__ float dstage[2][Mm];   // async-staged D anti-diagonals
  __shared__ float red[Mm];         // max reduction (COMBINE only)

  const int col = BACK ? (Mm - 1 - j) : j;         // physical column
  float lmax = NEGV;

  // Issue D element for diagonal k into slot k&1. Row clamped so every lane
  // of every wave issues on every step -> uniform ASYNCcnt bookkeeping.
  auto issue = [&](int k) {
    int i = k - j;
    i = i < 0 ? 0 : (i > Nn - 1 ? Nn - 1 : i);
    const int r = BACK ? (Nn - 1 - i) : i;         // physical row
    dp_stage_load(Db + r * Mm + col, &dstage[k & 1][j]);
  };

  issue(0);
  for (int k = 0; k < KD; ++k) {
    if (k + 1 < KD) {
      issue(k + 1);
#if DP_USE_ASYNC
      __builtin_amdgcn_s_wait_asynccnt(1);   // diag k staged; k+1 in flight
#endif
    } else {
#if DP_USE_ASYNC
      __builtin_amdgcn_s_wait_asynccnt(0);
#endif
    }

    const int i = k - j;
    if (i >= 0 && i < Nn) {
      const float d  = dstage[k & 1][j];
      // km1 = (k-1)%3 == (k+2)%3 ; km2 = (k-2)%3 == (k+1)%3
      const float dg = (i == 0 && j == 0) ? 0.0f
                     : ((i > 0 && j > 0) ? rbuf[(k + 1) % 3][j - 1] : NEGV);
      const float up = (i > 0) ? rbuf[(k + 2) % 3][j]     : NEGV;
      const float lf = (j > 0) ? rbuf[(k + 2) % 3][j - 1] : NEGV;
      const float m  = fmaxf(fmaxf(dg, up), lf);
      const float v  = d + m +
          __logf(__expf(dg - m) + __expf(up - m) + __expf(lf - m));
      rbuf[k % 3][j] = v;

      const int r   = BACK ? (Nn - 1 - i) : i;
      const int idx = r * Mm + col;
      if (!COMBINE) {
        Tb[idx] = v;                             // f (fwd) or raw b (bwd)
      } else {
        const float logit = Tb[idx] + v - d;     // f + b - d
        Tb[idx] = logit;
        lmax = fmaxf(lmax, logit);
      }
    }
    __syncthreads();   // publishes diagonal k before anyone touches k+1
  }

  if (COMBINE) {
    red[j] = lmax;
    __syncthreads();
    for (int s = Mm / 2; s > 0; s >>= 1) {
      if (j < s) red[j] = fmaxf(red[j], red[j + s]);
      __syncthreads();
    }
    if (j == 0) atomicMax(gmax, enc_f32(red[0]));
  }
}

// Sequential path (tiny workspace): fwd writes f to out, bwd combines in place.
__global__ __launch_bounds__(Mm) void dp_fwd_kernel(const float* __restrict__ D,
                                                    float* __restrict__ F) {
  dp_core<false, false>(D, F, nullptr, blockIdx.x);
}
__global__ __launch_bounds__(Mm) void dp_bwd_combine_kernel(const float* __restrict__ D,
                                                            float* __restrict__ F,
                                                            unsigned* __restrict__ gmax) {
  dp_core<true, true>(D, F, gmax, blockIdx.x);
}

// Fused path (big workspace): 2B concurrent blocks — fwd DPs fill out with f,
// bwd DPs fill ws with raw b. Doubles WGP utilization; ~halves wall time.
__global__ __launch_bounds__(Mm) void dp_both_kernel(const float* __restrict__ D,
                                                     float* __restrict__ F,
                                                     float* __restrict__ Bw,
                                                     int B) {
  const int blk = (int)blockIdx.x;
  if (blk < B) dp_core<false, false>(D, F,  nullptr, blk);
  else         dp_core<true,  false>(D, Bw, nullptr, blk - B);
}

// logit = f + b - d (float4, bandwidth-bound) + global max reduction.
__global__ __launch_bounds__(256) void combine_kernel(const float* __restrict__ D,
                                                      const float* __restrict__ Bw,
                                                      float* __restrict__ F,
                                                      unsigned* __restrict__ gmax,
                                                      int n4) {
  __shared__ float red[256];
  float lmax = NEGV;
  const float4* d4 = reinterpret_cast<const float4*>(D);
  const float4* b4 = reinterpret_cast<const float4*>(Bw);
  float4*       f4 = reinterpret_cast<float4*>(F);
  for (int idx = blockIdx.x * blockDim.x + threadIdx.x; idx < n4;
       idx += gridDim.x * blockDim.x) {
    const float4 f = f4[idx], b = b4[idx], d = d4[idx];
    float4 l;
    l.x = f.x + b.x - d.x;
    l.y = f.y + b.y - d.y;
    l.z = f.z + b.z - d.z;
    l.w = f.w + b.w - d.w;
    f4[idx] = l;
    lmax = fmaxf(lmax, fmaxf(fmaxf(l.x, l.y), fmaxf(l.z, l.w)));
  }
  const int t = threadIdx.x;
  red[t] = lmax;
  __syncthreads();
  for (int s = 128; s > 0; s >>= 1) {
    if (t < s) red[t] = fmaxf(red[t], red[t + s]);
    __syncthreads();
  }
  if (t == 0) atomicMax(gmax, enc_f32(red[0]));
}

__global__ void init_max_kernel(unsigned* gmax) {
  if (threadIdx.x == 0 && blockIdx.x == 0) *gmax = 0u;  // <= enc(-inf)
}

__global__ __launch_bounds__(256) void sub_max_kernel(float* __restrict__ out,
                                                      const unsigned* __restrict__ gmax,
                                                      int n4) {
  const float mx    = dec_f32(*gmax);
  const float scale = GAMMAc / Tc;
  float4* o = reinterpret_cast<float4*>(out);
  for (int idx = blockIdx.x * blockDim.x + threadIdx.x; idx < n4;
       idx += gridDim.x * blockDim.x) {
    float4 v = o[idx];
    v.x = scale * (v.x - mx);
    v.y = scale * (v.y - mx);
    v.z = scale * (v.z - mx);
    v.w = scale * (v.w - mx);
    o[idx] = v;
  }
}

extern "C" void kernel_launch(void* const* d_in, const int* in_sizes, int n_in,
                              void* d_out, int out_size, void* d_ws, size_t ws_size,
                              hipStream_t stream) {
  const float* d   = (const float*)d_in[0];
  float*       out = (float*)d_out;
  unsigned*    gmx = (unsigned*)d_ws;            // scalar max key at ws[0]

  const int B  = in_sizes[0] / (Nn * Mm);        // 64
  const int n4 = out_size / 4;

  init_max_kernel<<<1, 32, 0, stream>>>(gmx);

  const size_t need_fused = 256 + (size_t)B * Nn * Mm * sizeof(float);
  if (ws_size >= need_fused) {
    // Fused: 2B DP workgroups run concurrently (fwd -> out, bwd -> ws).
    float* bws = (float*)((char*)d_ws + 256);
    dp_both_kernel<<<2 * B, Mm, 0, stream>>>(d, out, bws, B);
    combine_kernel<<<2048, 256, 0, stream>>>(d, bws, out, gmx, n4);
  } else {
    // Sequential fallback: only 4 bytes of workspace required.
    dp_fwd_kernel<<<B, Mm, 0, stream>>>(d, out);
    dp_bwd_combine_kernel<<<B, Mm, 0, stream>>>(d, out, gmx);
  }

  // out = GAMMA * (logit - max) / T.
  sub_max_kernel<<<1024, 256, 0, stream>>>(out, gmx, n4);
}